// GINELayer_44813688766820
// MI455X (gfx1250) — compile-verified
//
#include <hip/hip_runtime.h>

#define DIM 128
#define WAVES_PER_BLOCK 8
#define BLOCK (WAVES_PER_BLOCK * 32)

typedef float v2f __attribute__((ext_vector_type(2)));
typedef float v8f __attribute__((ext_vector_type(8)));

// ---------------------------------------------------------------------------
// Kernel 1: h = relu(LayerNorm(x)); aggr = 0.  One wave32 per node.
// Lane l owns features [4l, 4l+4) -> one float4, fully coalesced per wave.
// ---------------------------------------------------------------------------
__global__ void __launch_bounds__(BLOCK)
ln_relu_zero_kernel(const float* __restrict__ x,
                    const float* __restrict__ gamma,
                    const float* __restrict__ beta,
                    float* __restrict__ h,
                    float* __restrict__ aggr,
                    int n_nodes) {
  int wave = (blockIdx.x * blockDim.x + threadIdx.x) >> 5;
  int lane = threadIdx.x & 31;
  if (wave >= n_nodes) return;

  const float4 v = ((const float4*)(x + (size_t)wave * DIM))[lane];
  float s  = v.x + v.y + v.z + v.w;
  float s2 = v.x * v.x + v.y * v.y + v.z * v.z + v.w * v.w;
  #pragma unroll
  for (int off = 16; off > 0; off >>= 1) {
    s  += __shfl_xor(s,  off, 32);
    s2 += __shfl_xor(s2, off, 32);
  }
  const float mu  = s * (1.0f / DIM);
  const float var = s2 * (1.0f / DIM) - mu * mu;
  const float rs  = rsqrtf(var + 1e-5f);

  const float4 g = ((const float4*)gamma)[lane];
  const float4 b = ((const float4*)beta)[lane];
  float4 hv;
  hv.x = fmaxf((v.x - mu) * rs * g.x + b.x, 0.0f);
  hv.y = fmaxf((v.y - mu) * rs * g.y + b.y, 0.0f);
  hv.z = fmaxf((v.z - mu) * rs * g.z + b.z, 0.0f);
  hv.w = fmaxf((v.w - mu) * rs * g.w + b.w, 0.0f);

  ((float4*)(h + (size_t)wave * DIM))[lane] = hv;
  const float4 z = {0.0f, 0.0f, 0.0f, 0.0f};
  ((float4*)(aggr + (size_t)wave * DIM))[lane] = z;
}

// ---------------------------------------------------------------------------
// Kernel 2: per-edge gather + atomic scatter-add.
// msg = relu(h[src]) == h[src] (h already ReLU'd).  One wave32 per edge:
// float4 coalesced gather, 4x global_atomic_add_f32 into aggr[dst].
// ---------------------------------------------------------------------------
__global__ void __launch_bounds__(BLOCK)
edge_scatter_kernel(const int* __restrict__ ei,
                    const float* __restrict__ h,
                    float* __restrict__ aggr,
                    int n_edges) {
  int wave = (blockIdx.x * blockDim.x + threadIdx.x) >> 5;
  int lane = threadIdx.x & 31;
  if (wave >= n_edges) return;

  const int src = ei[wave];
  const int dst = ei[n_edges + wave];

  const float4 m = ((const float4*)(h + (size_t)src * DIM))[lane];
  float* ap = aggr + (size_t)dst * DIM + lane * 4;
  atomicAdd(ap + 0, m.x);
  atomicAdd(ap + 1, m.y);
  atomicAdd(ap + 2, m.z);
  atomicAdd(ap + 3, m.w);
}

// ---------------------------------------------------------------------------
// Kernel 3: aggr = aggr + (1+eps)*h   (GEMM input matrix M, in place)
// ---------------------------------------------------------------------------
__global__ void __launch_bounds__(256)
fuse_kernel(const float* __restrict__ h,
            const float* __restrict__ eps_p,
            float* __restrict__ aggr,
            int n) {
  int i = blockIdx.x * blockDim.x + threadIdx.x;
  if (i >= n) return;
  const float scale = 1.0f + eps_p[0];
  aggr[i] = fmaf(scale, h[i], aggr[i]);
}

// ---------------------------------------------------------------------------
// Kernel 4: out = M @ W^T + bias + x  via V_WMMA_F32_16X16X4_F32.
// One wave per 16x16 output tile; K=128 in steps of 4 -> 32 WMMA / tile.
// A 16x4 f32 layout: lanes 0-15 hold K={0,1}, lanes 16-31 hold K={2,3},
// row M = lane&15.  B 4x16 mirrors it with N = lane&15; since
// out[n,o] = sum_k M[n,k]*W[o,k], B[k][n] = W[col0+n][k] -> identical
// addressing on W.  C/D: VGPR r holds row (r + 8*half), col = lane&15.
// N_NODES = 100000 = 16*6250 -> exact tiling, no divergence (EXEC all 1s).
// ---------------------------------------------------------------------------
__global__ void __launch_bounds__(BLOCK)
gemm_kernel(const float* __restrict__ Mmat,
            const float* __restrict__ W,
            const float* __restrict__ bias,
            const float* __restrict__ x,
            float* __restrict__ out,
            int n_tiles) {
  int wave = threadIdx.x >> 5;
  int lane = threadIdx.x & 31;
  int tile = blockIdx.x * WAVES_PER_BLOCK + wave;
  if (tile >= n_tiles) return;           // wave-uniform branch

  const int tm   = tile >> 3;            // row tile  (DIM/16 == 8 col tiles)
  const int tn   = tile & 7;             // col tile
  const int row0 = tm * 16;
  const int col0 = tn * 16;
  const int half = lane >> 4;            // 0: K lo pair, 1: K hi pair
  const int l    = lane & 15;

  const float* arow = Mmat + (size_t)(row0 + l) * DIM + 2 * half;
  const float* brow = W    + (size_t)(col0 + l) * DIM + 2 * half;

  v8f c = {};
  #pragma unroll
  for (int k = 0; k < DIM; k += 4) {
    v2f a = *(const v2f*)(arow + k);
    v2f b = *(const v2f*)(brow + k);
    c = __builtin_amdgcn_wmma_f32_16x16x4_f32(
        /*neg_a=*/false, a, /*neg_b=*/false, b,
        /*c_mod=*/(short)0, c, /*reuse_a=*/false, /*reuse_b=*/false);
  }

  const float bv = bias[col0 + l];
  #pragma unroll
  for (int r = 0; r < 8; ++r) {
    const int row = row0 + r + 8 * half;
    const size_t idx = (size_t)row * DIM + col0 + l;
    out[idx] = c[r] + bv + x[idx];
  }
}

// ---------------------------------------------------------------------------
extern "C" void kernel_launch(void* const* d_in, const int* in_sizes, int n_in,
                              void* d_out, int out_size, void* d_ws, size_t ws_size,
                              hipStream_t stream) {
  const float* x     = (const float*)d_in[0];
  const int*   ei    = (const int*)  d_in[1];
  const float* gamma = (const float*)d_in[2];
  const float* beta  = (const float*)d_in[3];
  const float* W     = (const float*)d_in[4];
  const float* bias  = (const float*)d_in[5];
  const float* eps   = (const float*)d_in[6];
  float* out = (float*)d_out;

  const int n_nodes = in_sizes[0] / DIM;
  const int n_edges = in_sizes[1] / 2;

  float* h    = (float*)d_ws;                       // [n_nodes * DIM]
  float* aggr = h + (size_t)n_nodes * DIM;          // [n_nodes * DIM]

  // 1) h = relu(LN(x)); aggr = 0
  {
    int blocks = (n_nodes + WAVES_PER_BLOCK - 1) / WAVES_PER_BLOCK;
    ln_relu_zero_kernel<<<blocks, BLOCK, 0, stream>>>(x, gamma, beta, h, aggr, n_nodes);
  }
  // 2) aggr[dst] += h[src]  over all edges
  {
    int blocks = (n_edges + WAVES_PER_BLOCK - 1) / WAVES_PER_BLOCK;
    edge_scatter_kernel<<<blocks, BLOCK, 0, stream>>>(ei, h, aggr, n_edges);
  }
  // 3) aggr += (1+eps)*h
  {
    int n = n_nodes * DIM;
    int blocks = (n + 255) / 256;
    fuse_kernel<<<blocks, 256, 0, stream>>>(h, eps, aggr, n);
  }
  // 4) out = aggr @ W^T + bias + x   (WMMA f32 16x16x4)
  {
    int n_tiles = (n_nodes / 16) * (DIM / 16);
    int blocks = (n_tiles + WAVES_PER_BLOCK - 1) / WAVES_PER_BLOCK;
    gemm_kernel<<<blocks, BLOCK, 0, stream>>>(aggr, W, bias, x, out, n_tiles);
  }
}